// SkinModel_RotationOnly_87076166959883
// MI455X (gfx1250) — compile-verified
//
#include <hip/hip_runtime.h>
#include <hip/hip_bf16.h>

// ---------------------------------------------------------------------------
// SMPL rotation-only skinning for MI455X (gfx1250, wave32).
//
// out[b,i,n] = sum_{k,j} M_k[b][i][j] * w[b,k,n] * nrm[b,j,n]
//            = A[b](3x72) @ F[b](72xN),  A[i][3k+j]=M_k[i][j], F[3k+j]=w_k*n_j
//
// Memory-bound: ~192 MB streamed once -> ~8 us floor at 23.3 TB/s.
// GEMM per 16-vertex tile done with 18x V_WMMA_F32_16X16X4_F32 (fp32, matches
// reference precision). A is padded 16x72 in LDS; B operand built in-lane.
// ---------------------------------------------------------------------------

typedef __attribute__((ext_vector_type(2))) float v2f;
typedef __attribute__((ext_vector_type(8))) float v8f;

#define NJ 24

__constant__ int SMPL_PAR[NJ] = {-1, 0, 0, 0, 1, 2, 3, 4, 5, 6, 7, 8,
                                 9, 9, 9, 12, 13, 14, 16, 17, 18, 19, 20, 21};

// ---- Stage 1: Rodrigues + kinematic chain -> padded A[b] (16 x 72) ---------
__global__ void build_A_kernel(const float* __restrict__ pose,
                               float* __restrict__ Aout, int B) {
  int b = blockIdx.x * blockDim.x + threadIdx.x;
  if (b >= B) return;

  float R[NJ][9];
  for (int k = 0; k < NJ; ++k) {
    float ax = pose[b * 72 + 3 * k + 0];
    float ay = pose[b * 72 + 3 * k + 1];
    float az = pose[b * 72 + 3 * k + 2];
    // reference: angle = ||axisang + 1e-8|| (eps added per component)
    float tx = ax + 1e-8f, ty = ay + 1e-8f, tz = az + 1e-8f;
    float ang = sqrtf(tx * tx + ty * ty + tz * tz);
    float inv = 1.0f / ang;
    float ux = ax * inv, uy = ay * inv, uz = az * inv;  // axis = axisang/angle
    float co = cosf(ang), si = sinf(ang), t = 1.0f - co;
    float M[9];
    M[0] = co + t * ux * ux;  M[1] = t * ux * uy - si * uz;  M[2] = t * ux * uz + si * uy;
    M[3] = t * uy * ux + si * uz;  M[4] = co + t * uy * uy;  M[5] = t * uy * uz - si * ux;
    M[6] = t * uz * ux - si * uy;  M[7] = t * uz * uy + si * ux;  M[8] = co + t * uz * uz;
    int p = SMPL_PAR[k];
    if (p < 0) {
      for (int q = 0; q < 9; ++q) R[k][q] = M[q];
    } else {
      for (int i = 0; i < 3; ++i)
        for (int j = 0; j < 3; ++j) {
          float s = 0.0f;
          for (int l = 0; l < 3; ++l) s += R[p][3 * i + l] * M[3 * l + j];
          R[k][3 * i + j] = s;
        }
    }
  }
  // Padded A: (B, 16, 72); rows 3..15 zero so WMMA garbage rows are harmless.
  float* Ab = Aout + (size_t)b * 16 * 72;
  for (int m = 0; m < 16; ++m)
    for (int kk = 0; kk < 72; ++kk) {
      float v = 0.0f;
      if (m < 3) v = R[kk / 3][m * 3 + (kk % 3)];
      Ab[m * 72 + kk] = v;
    }
}

// ---- Stage 2: streaming skinning via V_WMMA_F32_16X16X4_F32 ---------------
__global__ void __launch_bounds__(256)
skin_wmma_kernel(const float* __restrict__ normals,   // (B,3,N)
                 const float* __restrict__ weights,   // (B,24,N)
                 const float* __restrict__ Amat,      // (B,16,72)
                 float* __restrict__ out,             // (B,3,N)
                 int N) {
  __shared__ float Alds[16 * 72];
  const int b = blockIdx.y;
  for (int i = threadIdx.x; i < 16 * 72; i += blockDim.x)
    Alds[i] = Amat[(size_t)b * 16 * 72 + i];
  __syncthreads();

  const int lane  = threadIdx.x & 31;
  const int col16 = lane & 15;   // B-operand column / A-operand row
  const int hl    = lane >> 4;   // lane-half selects K pair {0,1} vs {2,3}
  const int wavesPerBlock = blockDim.x >> 5;
  const int waveId     = blockIdx.x * wavesPerBlock + (threadIdx.x >> 5);
  const int waveStride = gridDim.x * wavesPerBlock;
  const int numTiles   = (N + 15) >> 4;

  const float* Wb = weights + (size_t)b * NJ * N;
  const float* Nb = normals + (size_t)b * 3 * N;
  float*       Ob = out     + (size_t)b * 3 * N;
  // A layout in LDS: row-major 16x72. Lane needs A[row][4c + 2*hl + {0,1}]
  // per chunk -> contiguous pair -> ds_load_b64.
  const float* Ap = &Alds[col16 * 72 + 2 * hl];

  for (int t = waveId; t < numTiles; t += waveStride) {
    const int col  = t * 16 + col16;
    const int colc = (col < N) ? col : (N - 1);   // clamp (tail safety)

    // Single-pass streams: non-temporal. Lanes 16-31 mirror lanes 0-15's
    // addresses (same cachelines; avoids cross-lane shuffles for B operand).
    float w[NJ];
#pragma unroll
    for (int k = 0; k < NJ; ++k)
      w[k] = __builtin_nontemporal_load(&Wb[(size_t)k * N + colc]);
    float nr[3];
#pragma unroll
    for (int j = 0; j < 3; ++j)
      nr[j] = __builtin_nontemporal_load(&Nb[(size_t)j * N + colc]);

    v8f acc = {0.f, 0.f, 0.f, 0.f, 0.f, 0.f, 0.f, 0.f};
#pragma unroll
    for (int c = 0; c < 18; ++c) {               // K = 72 in chunks of 4
      const int kj = 4 * c;                      // compile-time F indices
      float f0 = w[(kj + 0) / 3] * nr[(kj + 0) % 3];
      float f1 = w[(kj + 1) / 3] * nr[(kj + 1) % 3];
      float f2 = w[(kj + 2) / 3] * nr[(kj + 2) % 3];
      float f3 = w[(kj + 3) / 3] * nr[(kj + 3) % 3];
      v2f bv;                                     // B 4x16: v0:K=0/2, v1:K=1/3
      bv.x = hl ? f2 : f0;
      bv.y = hl ? f3 : f1;
      v2f av = *(const v2f*)(Ap + 4 * c);         // A 16x4: v0:K=0/2, v1:K=1/3
      acc = __builtin_amdgcn_wmma_f32_16x16x4_f32(
          false, av, false, bv, (short)0, acc, false, false);
    }
    // D rows 0..2 live in VGPRs 0..2 of lanes 0..15.
    if (hl == 0 && col < N) {
      __builtin_nontemporal_store(acc[0], &Ob[(size_t)0 * N + col]);
      __builtin_nontemporal_store(acc[1], &Ob[(size_t)1 * N + col]);
      __builtin_nontemporal_store(acc[2], &Ob[(size_t)2 * N + col]);
    }
  }
}

extern "C" void kernel_launch(void* const* d_in, const int* in_sizes, int n_in,
                              void* d_out, int out_size, void* d_ws, size_t ws_size,
                              hipStream_t stream) {
  const float* normals = (const float*)d_in[0];  // (B,3,N)
  const float* pose    = (const float*)d_in[1];  // (B,72)
  const float* weights = (const float*)d_in[2];  // (B,24,N)
  float* out = (float*)d_out;

  const int B = in_sizes[1] / 72;          // 16
  const int N = in_sizes[0] / (3 * B);     // 100000
  float* Amat = (float*)d_ws;              // B*16*72 floats = 72 KB

  build_A_kernel<<<dim3((B + 31) / 32), dim3(32), 0, stream>>>(pose, Amat, B);

  dim3 grid(128, B);                        // 8 wave32 per block, grid-stride
  skin_wmma_kernel<<<grid, dim3(256), 0, stream>>>(normals, weights, Amat, out, N);
}